// GeneGNN_9929964389195
// MI455X (gfx1250) — compile-verified
//
#include <hip/hip_runtime.h>

typedef float v2f __attribute__((ext_vector_type(2)));
typedef float v8f __attribute__((ext_vector_type(8)));

#define NB     32
#define NN     2000
#define NE     32000
#define HD     640
#define OD     1280
#define NTOT   (NB * NN)
#define NCHUNK 16
#define ECH    (NE / NCHUNK)   // 2000 edges per chunk
#define NPAD   2048            // padded node count for partial arrays

// ---------------------------------------------------------------------------
// Workspace layout (float units):
//   [0      , 2048  )  dinv   : deg^{-1/2}, nodes 0..1999 (pad to 2048)
//   [2048   , 4096  )  wnode  : batch-0 node weight c_0[v]
//   [4096   , 68096 )  s      : per-node scalar aggregation (64000 nodes)
//   [68096  , 88576 )  H      : [32 x 640] reduced features
//   [88576  , 121344)  pdeg   : int partial degree   [NCHUNK][NPAD]
//   [121344 , 154112)  ps     : float partial s      [NCHUNK][NPAD]
//   [154112 , 186880)  pw     : float partial wnode  [NCHUNK][NPAD]
// ---------------------------------------------------------------------------

// Pass 1a: partial in-degree per (node, edge-chunk). grid (8, NCHUNK) x 256.
__global__ void k_deg_part(const int* __restrict__ ei, int* __restrict__ pdeg) {
    __shared__ int sh_dst[256];
    const int v     = blockIdx.x * 256 + threadIdx.x;     // 0..2047
    const int ebase = blockIdx.y * ECH;
    const int eend  = ebase + ECH;
    int cnt = 0;
    for (int e0 = ebase; e0 < eend; e0 += 256) {
        const int e = e0 + threadIdx.x;
        sh_dst[threadIdx.x] = (e < eend) ? ei[NE + e] : -1;
        __syncthreads();
#pragma unroll 8
        for (int i = 0; i < 256; ++i) cnt += (sh_dst[i] == v);
        __syncthreads();
    }
    pdeg[blockIdx.y * NPAD + v] = cnt;
}

// Pass 1b: combine partial degrees in fixed chunk order -> dinv.
__global__ void k_deg_comb(const int* __restrict__ pdeg, float* __restrict__ dinv) {
    const int v = blockIdx.x * 256 + threadIdx.x;         // 0..2047
    int cnt = 1;                                          // self-loop
#pragma unroll
    for (int c = 0; c < NCHUNK; ++c) cnt += pdeg[c * NPAD + v];
    dinv[v] = rsqrtf((float)cnt);                         // v>=2000: deg==1 -> 1.0
}

// Pass 2a: partial s / wnode per (node, edge-chunk). grid (8, NCHUNK) x 256.
__global__ void k_swn_part(const int* __restrict__ ei, const float* __restrict__ x,
                           const float* __restrict__ dinv,
                           float* __restrict__ ps, float* __restrict__ pw) {
    __shared__ int sh_src[256];
    __shared__ int sh_dst[256];
    const int v     = blockIdx.x * 256 + threadIdx.x;     // 0..2047
    const int ebase = blockIdx.y * ECH;
    const int eend  = ebase + ECH;
    const float dv  = dinv[v];
    float sv = 0.0f, wv = 0.0f;
    for (int e0 = ebase; e0 < eend; e0 += 256) {
        const int e = e0 + threadIdx.x;
        const bool ok = (e < eend);
        sh_src[threadIdx.x] = ok ? ei[e]      : -1;
        sh_dst[threadIdx.x] = ok ? ei[NE + e] : -1;
        __syncthreads();
        for (int i = 0; i < 256; ++i) {
            const int sr = sh_src[i];
            const int ds = sh_dst[i];
            if (ds == v) sv += x[sr] * dinv[sr] * dv;     // message into v
            if (sr == v) wv += dv * dinv[ds];             // v feeds batch-0 outputs
        }
        __syncthreads();
    }
    ps[blockIdx.y * NPAD + v] = sv;
    pw[blockIdx.y * NPAD + v] = wv;
}

// Pass 2b: combine partials (fixed order) + self-loop terms; fill s for all
// 64000 nodes. grid 250 x 256.
__global__ void k_swn_comb(const float* __restrict__ x, const float* __restrict__ dinv,
                           const float* __restrict__ ps, const float* __restrict__ pw,
                           float* __restrict__ s, float* __restrict__ wnode) {
    const int v = blockIdx.x * 256 + threadIdx.x;
    if (v < NN) {
        const float dv = dinv[v];
        float sv = x[v] * dv * dv;                        // self-loop message
        float wv = dv * dv;                               // self-loop weight
#pragma unroll
        for (int c = 0; c < NCHUNK; ++c) {
            sv += ps[c * NPAD + v];
            wv += pw[c * NPAD + v];
        }
        s[v] = sv;
        wnode[v] = wv;
    } else if (v < NTOT) {
        s[v] = x[v];                                      // deg==1 -> norm==1
    }
}

// Pass 3: H[b][j] = sum_n w(b,n) * relu(s[b*NN+n]*W1[j] + b1[j])
// grid (5, 32) x 128 -> j = bx*128+tid covers 640, one batch per by.
__global__ void k_reduceH(const float* __restrict__ s,
                          const float* __restrict__ wnode,
                          const float* __restrict__ W1,
                          const float* __restrict__ b1,
                          float* __restrict__ Hbuf) {
    const int b = blockIdx.y;
    const int j = blockIdx.x * 128 + threadIdx.x;         // < 640
    const float w1j = W1[j];
    const float b1j = b1[j];
    const float* sb = s + b * NN;
    __shared__ float sh_s[128];
    __shared__ float sh_w[128];
    float acc = 0.0f;
    for (int n0 = 0; n0 < NN; n0 += 128) {
        const int t  = n0 + threadIdx.x;
        const bool ok = (t < NN);
        sh_s[threadIdx.x] = ok ? sb[t] : 0.0f;
        sh_w[threadIdx.x] = ok ? ((b == 0) ? wnode[t] : 1.0f) : 0.0f;
        __syncthreads();
        const int lim = (NN - n0 < 128) ? (NN - n0) : 128;
        for (int i = 0; i < lim; ++i)
            acc += sh_w[i] * fmaxf(fmaf(sh_s[i], w1j, b1j), 0.0f);
        __syncthreads();
    }
    Hbuf[b * HD + j] = acc;
}

// Pass 4: OUT = (1/N) * H @ W2 + b2 via V_WMMA_F32_16X16X4_F32 (full f32).
// One wave per 16x16 output tile; grid (80, 2) x 32.
// A 16x4 f32 layout: lanes 0-15 hold K0/K1 in v0/v1; lanes 16-31 hold K2/K3.
// B 4x16 mirrors with N striped across lanes. C/D: vgpr r -> M = r + 8*(lane>=16).
__global__ void k_gemm_wmma(const float* __restrict__ Hbuf,
                            const float* __restrict__ W2,
                            const float* __restrict__ b2,
                            float* __restrict__ out) {
    const int lane  = threadIdx.x;
    const int l     = lane & 15;
    const int g     = lane >> 4;
    const int mtile = blockIdx.y;
    const int ntile = blockIdx.x;
    const int col   = ntile * 16 + l;
    const float* arow = Hbuf + (mtile * 16 + l) * HD;

    v8f acc = {};
    for (int k0 = 0; k0 < HD; k0 += 4) {
        const int kk = k0 + 2 * g;
        v2f a = *(const v2f*)(arow + kk);                 // K=kk, kk+1 for this lane
        v2f bb;
        bb.x = W2[kk * OD + col];
        bb.y = W2[(kk + 1) * OD + col];
        acc = __builtin_amdgcn_wmma_f32_16x16x4_f32(
            /*neg_a=*/false, a, /*neg_b=*/false, bb,
            /*c_mod=*/(short)0, acc, /*reuse_a=*/false, /*reuse_b=*/false);
    }

    const float bias = b2[col];
#pragma unroll
    for (int r = 0; r < 8; ++r) {
        const int row = mtile * 16 + g * 8 + r;
        out[row * OD + col] = acc[r] * (1.0f / (float)NN) + bias;
    }
}

extern "C" void kernel_launch(void* const* d_in, const int* in_sizes, int n_in,
                              void* d_out, int out_size, void* d_ws, size_t ws_size,
                              hipStream_t stream) {
    (void)in_sizes; (void)n_in; (void)out_size; (void)ws_size;
    const float* x  = (const float*)d_in[0];   // [32, 2000]
    const int*   ei = (const int*)d_in[1];     // [2, 32000]
    const float* W1 = (const float*)d_in[2];   // [1, 640]
    const float* b1 = (const float*)d_in[3];   // [640]
    const float* W2 = (const float*)d_in[4];   // [640, 1280]
    const float* b2 = (const float*)d_in[5];   // [1280]
    float* outp = (float*)d_out;               // [32, 1280]

    float* wf    = (float*)d_ws;
    float* dinv  = wf;
    float* wnode = wf + 2048;
    float* s     = wf + 4096;
    float* Hbuf  = wf + 68096;
    int*   pdeg  = (int*)(wf + 88576);
    float* ps    = wf + 121344;
    float* pw    = wf + 154112;

    k_deg_part<<<dim3(NPAD / 256, NCHUNK), dim3(256), 0, stream>>>(ei, pdeg);
    k_deg_comb<<<dim3(NPAD / 256), dim3(256), 0, stream>>>(pdeg, dinv);
    k_swn_part<<<dim3(NPAD / 256, NCHUNK), dim3(256), 0, stream>>>(ei, x, dinv, ps, pw);
    k_swn_comb<<<dim3((NTOT + 255) / 256), dim3(256), 0, stream>>>(x, dinv, ps, pw, s, wnode);
    k_reduceH<<<dim3(5, 32), dim3(128), 0, stream>>>(s, wnode, W1, b1, Hbuf);
    k_gemm_wmma<<<dim3(OD / 16, NB / 16), dim3(32), 0, stream>>>(Hbuf, W2, b2, outp);
}